// Zeroshot_50749333570318
// MI455X (gfx1250) — compile-verified
//
#include <hip/hip_runtime.h>
#include <hip/hip_bf16.h>
#include <math.h>

typedef float v2f __attribute__((ext_vector_type(2)));
typedef float v8f __attribute__((ext_vector_type(8)));

#define GEMM_WAVES 4   // waves per block
#define GEMM_NT    4   // 16-col tiles per wave: 4 independent WMMA chains, A reused x4

// out[m,n] = act( sum_k A[m,k] * W[n,k] + bias[n] )   (all fp32)
// A: MxK row-major (lda), W: NxK row-major (ldw), out: MxN (ldo).
// Core op: v_wmma_f32_16x16x4_f32  D(16x16) += A(16x4) x B(4x16), wave32.
// Software-pipelined K loop: operands for k+4 are in flight while WMMAs for k issue.
__global__ __launch_bounds__(GEMM_WAVES * 32)
void gemm_nt_wmma_f32(const float* __restrict__ A, int lda,
                      const float* __restrict__ W, int ldw,
                      const float* __restrict__ bias,
                      float* __restrict__ out, int ldo,
                      int M, int N, int K, int do_relu)
{
    const int lane = threadIdx.x & 31;
    const int wave = threadIdx.x >> 5;
    const int half = lane >> 4;     // which K-pair this lane holds (k+0,1 vs k+2,3)
    const int idx  = lane & 15;     // A row / B col within the 16x16 tile

    const int m0 = blockIdx.y * 16;
    const int n0 = (blockIdx.x * GEMM_WAVES + wave) * (16 * GEMM_NT);
    if (n0 >= N) return;            // wave-uniform early out (EXEC stays full)

    // A: lane holds row m0+idx, elements k = k0 + 2*half + {0,1}  (ISA 32-bit A 16x4 layout)
    const int  arow   = m0 + idx;
    const bool avalid = arow < M;                       // clamp addr, mask store later
    const float* ap = A + (avalid ? arow : 0) * lda;

    // B: lane holds col n+idx, same per-lane K offsets (ISA B 4x16 layout)
    int          bcol[GEMM_NT];
    bool         bv[GEMM_NT];
    const float* bp[GEMM_NT];
    #pragma unroll
    for (int t = 0; t < GEMM_NT; ++t) {
        bcol[t] = n0 + 16 * t + idx;
        bv[t]   = bcol[t] < N;
        bp[t]   = W + (bv[t] ? bcol[t] : 0) * ldw;
    }

    v8f acc[GEMM_NT];
    #pragma unroll
    for (int t = 0; t < GEMM_NT; ++t)
        acc[t] = (v8f){0.f, 0.f, 0.f, 0.f, 0.f, 0.f, 0.f, 0.f};

    const int koff = 2 * half;

    // ---- software pipeline: preload k=0 fragment ----
    v2f a = *(const v2f*)(ap + koff);
    v2f b[GEMM_NT];
    #pragma unroll
    for (int t = 0; t < GEMM_NT; ++t)
        b[t] = *(const v2f*)(bp[t] + koff);

    for (int k = 4; k < K; k += 4) {
        // issue next fragment's loads before consuming the current one
        v2f an = *(const v2f*)(ap + k + koff);
        v2f bn[GEMM_NT];
        #pragma unroll
        for (int t = 0; t < GEMM_NT; ++t)
            bn[t] = *(const v2f*)(bp[t] + k + koff);

        #pragma unroll
        for (int t = 0; t < GEMM_NT; ++t)
            acc[t] = __builtin_amdgcn_wmma_f32_16x16x4_f32(
                         false, a, false, b[t], (short)0, acc[t], false, false);

        a = an;
        #pragma unroll
        for (int t = 0; t < GEMM_NT; ++t)
            b[t] = bn[t];
    }
    // drain
    #pragma unroll
    for (int t = 0; t < GEMM_NT; ++t)
        acc[t] = __builtin_amdgcn_wmma_f32_16x16x4_f32(
                     false, a, false, b[t], (short)0, acc[t], false, false);

    // D layout: VGPR r -> row m0 + r + 8*half, col = tile_col + idx
    const int rbase = m0 + 8 * half;
    #pragma unroll
    for (int t = 0; t < GEMM_NT; ++t) {
        if (bv[t]) {
            const float bb = bias ? bias[bcol[t]] : 0.f;
            #pragma unroll
            for (int r = 0; r < 8; ++r) {
                const int row = rbase + r;
                if (row < M) {
                    float v = acc[t][r] + bb;
                    if (do_relu) v = fmaxf(v, 0.f);
                    out[row * ldo + bcol[t]] = v;
                }
            }
        }
    }
}

#define H_MAX 900

// One block per (b,s). score = sigmoid(sum_h relu(hx+ha+b3)*W4 + b4),
// loss += (label - score)^2 / (B*S)
__global__ __launch_bounds__(256)
void score_loss_kernel(const float* __restrict__ hx,        // (B*S, H)
                       const float* __restrict__ ha,        // (B*N, H)
                       const float* __restrict__ b3,        // (H)
                       const float* __restrict__ W4,        // (H)
                       const float* __restrict__ b4,        // (1)
                       const int*   __restrict__ x_label,   // (B*S)
                       const int*   __restrict__ att_label, // (B*N)
                       float* __restrict__ loss,
                       int S, int N, int H, float inv_bs)
{
    __shared__ float u[H_MAX];    // hx[b,s,:] + b3
    __shared__ float w4s[H_MAX];

    const int bs   = blockIdx.x;
    const int b    = bs / S;
    const int tid  = threadIdx.x;
    const int lane = tid & 31;
    const int wave = tid >> 5;
    const int nwaves = blockDim.x >> 5;

    const float* hxp = hx + bs * H;
    for (int h = tid; h < H; h += blockDim.x) {
        u[h]   = hxp[h] + b3[h];
        w4s[h] = W4[h];
    }
    __syncthreads();

    const int   xl    = x_label[bs];
    const float bias4 = b4[0];
    float part = 0.f;

    for (int n = wave; n < N; n += nwaves) {
        const float* hap = ha + (b * N + n) * H;
        float acc = 0.f;
        for (int h = lane; h < H; h += 32)
            acc += fmaxf(u[h] + hap[h], 0.f) * w4s[h];
        // wave32 butterfly reduction
        #pragma unroll
        for (int off = 16; off > 0; off >>= 1)
            acc += __shfl_xor(acc, off, 32);
        if (lane == 0) {
            const float score = 1.f / (1.f + expf(-(acc + bias4)));
            const float label = (xl == att_label[b * N + n]) ? 1.f : 0.f;
            const float d = label - score;
            part += d * d;
        }
    }
    if (lane == 0)
        atomicAdd(loss, part * inv_bs);
}

__global__ void zero_scalar(float* p)
{
    if (threadIdx.x == 0 && blockIdx.x == 0) *p = 0.f;
}

extern "C" void kernel_launch(void* const* d_in, const int* in_sizes, int n_in,
                              void* d_out, int out_size, void* d_ws, size_t ws_size,
                              hipStream_t stream)
{
    (void)in_sizes; (void)n_in; (void)out_size; (void)ws_size;

    const float* x   = (const float*)d_in[0];
    const int*   xl  = (const int*)  d_in[1];
    const float* att = (const float*)d_in[2];
    const int*   al  = (const int*)  d_in[3];
    const float* W1  = (const float*)d_in[4];
    const float* b1  = (const float*)d_in[5];
    const float* W2  = (const float*)d_in[6];
    const float* b2  = (const float*)d_in[7];
    const float* W3  = (const float*)d_in[8];
    const float* b3  = (const float*)d_in[9];
    const float* W4  = (const float*)d_in[10];
    const float* b4  = (const float*)d_in[11];
    float* out = (float*)d_out;

    constexpr int B = 4, S = 128, C = 4096, N = 50;
    constexpr int AR = 312, H = 900;
    constexpr int BN = B * N;   // 200
    constexpr int BS = B * S;   // 512

    // fp32 workspace: 6.6 MB total
    float* ws   = (float*)d_ws;
    float* h1   = ws;                 // BN x H
    float* attf = h1   + BN * H;      // BN x C
    float* hx   = attf + BN * C;      // BS x H
    float* ha   = hx   + BS * H;      // BN x H

    const dim3 blk(GEMM_WAVES * 32);
    auto ntx = [](int n) { return (n + 16 * GEMM_NT * GEMM_WAVES - 1) /
                                  (16 * GEMM_NT * GEMM_WAVES); };
    auto mty = [](int m) { return (m + 15) / 16; };

    // h1 = relu(att @ W1^T + b1)            [200 x 900,  K=312]
    gemm_nt_wmma_f32<<<dim3(ntx(H), mty(BN)), blk, 0, stream>>>(
        att, AR, W1, AR, b1, h1, H, BN, H, AR, 1);
    // attf = relu(h1 @ W2^T + b2)           [200 x 4096, K=900]
    gemm_nt_wmma_f32<<<dim3(ntx(C), mty(BN)), blk, 0, stream>>>(
        h1, H, W2, H, b2, attf, C, BN, C, H, 1);
    // hx = x @ W3[:, :C]^T                  [512 x 900,  K=4096]
    gemm_nt_wmma_f32<<<dim3(ntx(H), mty(BS)), blk, 0, stream>>>(
        x, C, W3, 2 * C, (const float*)nullptr, hx, H, BS, H, C, 0);
    // ha = attf @ W3[:, C:]^T               [200 x 900,  K=4096]
    gemm_nt_wmma_f32<<<dim3(ntx(H), mty(BN)), blk, 0, stream>>>(
        attf, C, W3 + C, 2 * C, (const float*)nullptr, ha, H, BN, H, C, 0);

    // loss accumulator must be zeroed (harness poisons d_out)
    zero_scalar<<<dim3(1), dim3(32), 0, stream>>>(out);
    score_loss_kernel<<<dim3(BS), dim3(256), 0, stream>>>(
        hx, ha, b3, W4, b4, xl, al, out, S, N, H, 1.0f / (float)BS);
}